// Sender_54485955117384
// MI455X (gfx1250) — compile-verified
//
#include <hip/hip_runtime.h>
#include <hip/hip_bf16.h>

// ---------------- problem constants ----------------
#define N_GRAPHS 4096
#define NODES_PER_GRAPH 32
#define NNODES (N_GRAPHS * NODES_PER_GRAPH)   // 131072
#define NEDGES 1048576
#define F_IN 128
#define HEADS 4
#define D_HEAD 64
#define EMB (HEADS * D_HEAD)                  // 256
#define HIDDEN 256
#define NEG_SLOPE 0.2f

typedef float v2f __attribute__((ext_vector_type(2)));
typedef float v8f __attribute__((ext_vector_type(8)));

// =====================================================================
// K1: h = x @ W   (fp32 WMMA 16x16x4). Block = 256 thr = 8 waves.
// Each wave: 16 rows x 64 cols (4 accumulators); block: 32 rows x 256 cols.
// =====================================================================
__global__ __launch_bounds__(256) void gemm_xw_kernel(
    const float* __restrict__ X, const float* __restrict__ W,
    float* __restrict__ H) {
  const int tid  = threadIdx.x;
  const int wave = tid >> 5, lane = tid & 31;
  const int rowTile = blockIdx.x * 2 + (wave >> 2);  // 2 row tiles / block
  const int colQuad = wave & 3;                      // 4 x 64 cols
  const int row0 = rowTile * 16;
  const int col0 = colQuad * 64;
  const int half = lane >> 4;       // 0: K pair 0,1   1: K pair 2,3
  const int l15  = lane & 15;
  const int arow = row0 + l15;

  v8f acc0 = {}, acc1 = {}, acc2 = {}, acc3 = {};
  for (int k0 = 0; k0 < F_IN; k0 += 4) {
    const int ka = k0 + half * 2;
    v2f a;                                   // A 16x4: lanes0-15 K=k0,k0+1; lanes16-31 K=k0+2,k0+3
    a.x = X[arow * F_IN + ka];
    a.y = X[arow * F_IN + ka + 1];
    const float* Wk = W + ka * EMB;          // B 4x16: same half-lane K split
    v2f b0, b1, b2, b3;
    int c;
    c = col0 +  0 + l15; b0.x = Wk[c]; b0.y = Wk[EMB + c];
    c = col0 + 16 + l15; b1.x = Wk[c]; b1.y = Wk[EMB + c];
    c = col0 + 32 + l15; b2.x = Wk[c]; b2.y = Wk[EMB + c];
    c = col0 + 48 + l15; b3.x = Wk[c]; b3.y = Wk[EMB + c];
    acc0 = __builtin_amdgcn_wmma_f32_16x16x4_f32(false, a, false, b0, (short)0, acc0, false, false);
    acc1 = __builtin_amdgcn_wmma_f32_16x16x4_f32(false, a, false, b1, (short)0, acc1, false, false);
    acc2 = __builtin_amdgcn_wmma_f32_16x16x4_f32(false, a, false, b2, (short)0, acc2, false, false);
    acc3 = __builtin_amdgcn_wmma_f32_16x16x4_f32(false, a, false, b3, (short)0, acc3, false, false);
  }
  // C/D layout: VGPR r -> M=r (lanes0-15) / M=r+8 (lanes16-31), N = lane&15
  const int srow0 = row0 + half * 8;
#pragma unroll
  for (int r = 0; r < 8; ++r) {
    float* out = H + (srow0 + r) * EMB + l15;
    out[col0 +  0] = acc0[r];
    out[col0 + 16] = acc1[r];
    out[col0 + 32] = acc2[r];
    out[col0 + 48] = acc3[r];
  }
}

// =====================================================================
// K2: alpha_src[n,h] = h[n,h,:].a_src[h]   alpha_dst likewise
// =====================================================================
__global__ void alphas_kernel(const float* __restrict__ H,
                              const float* __restrict__ a_src,
                              const float* __restrict__ a_dst,
                              float* __restrict__ As, float* __restrict__ Ad) {
  int t = blockIdx.x * blockDim.x + threadIdx.x;
  if (t >= NNODES * HEADS) return;
  int n = t >> 2, hd = t & 3;
  const float* hp = H + n * EMB + hd * D_HEAD;
  const float* sp = a_src + hd * D_HEAD;
  const float* dp = a_dst + hd * D_HEAD;
  float ss = 0.f, dd = 0.f;
#pragma unroll 8
  for (int i = 0; i < D_HEAD; ++i) {
    float v = hp[i];
    ss += v * sp[i];
    dd += v * dp[i];
  }
  As[t] = ss;
  Ad[t] = dd;
}

// =====================================================================
// K3a: histogram of edges whose dst is the target or ego node of its graph
// =====================================================================
__global__ void edge_hist_kernel(const int* __restrict__ ei,
                                 const int* __restrict__ tgt,
                                 const int* __restrict__ ego,
                                 int* __restrict__ cnt) {
  int e = blockIdx.x * blockDim.x + threadIdx.x;
  if (e >= NEDGES) return;
  int dst = ei[NEDGES + e];
  int g = dst >> 5, dl = dst & 31;
  if (dl == tgt[g] || dl == ego[g]) atomicAdd(&cnt[g], 1);
}

// =====================================================================
// K3b: exclusive scan of 4096 counts (single block of 256 threads)
// =====================================================================
__global__ __launch_bounds__(256) void scan_kernel(const int* __restrict__ cnt,
                                                   int* __restrict__ off,
                                                   int* __restrict__ cursor) {
  __shared__ int tots[256];
  int t = threadIdx.x;
  int local[16];
  int s = 0;
#pragma unroll
  for (int i = 0; i < 16; ++i) { local[i] = s; s += cnt[t * 16 + i]; }
  tots[t] = s;
  __syncthreads();
  for (int d = 1; d < 256; d <<= 1) {
    int v = (t >= d) ? tots[t - d] : 0;
    __syncthreads();
    tots[t] += v;
    __syncthreads();
  }
  int base = (t == 0) ? 0 : tots[t - 1];
#pragma unroll
  for (int i = 0; i < 16; ++i) {
    int o = base + local[i];
    off[t * 16 + i] = o;
    cursor[t * 16 + i] = o;
  }
  if (t == 255) off[N_GRAPHS] = tots[255];
}

// =====================================================================
// K3c: scatter filtered edges (packed: src_local | dst_local<<8)
// =====================================================================
__global__ void edge_scatter_kernel(const int* __restrict__ ei,
                                    const int* __restrict__ tgt,
                                    const int* __restrict__ ego,
                                    int* __restrict__ cursor,
                                    int* __restrict__ ebuf) {
  int e = blockIdx.x * blockDim.x + threadIdx.x;
  if (e >= NEDGES) return;
  int dst = ei[NEDGES + e];
  int g = dst >> 5, dl = dst & 31;
  if (dl == tgt[g] || dl == ego[g]) {
    int sl = ei[e] & 31;                 // src in same graph by construction
    int pos = atomicAdd(&cursor[g], 1);
    ebuf[pos] = sl | (dl << 8);
  }
}

// =====================================================================
// K4: per-graph softmax + aggregation for the 2 interesting dst nodes,
// writes feat[g] = concat(h'[tgt], h'[ego]) + gat_bias
// =====================================================================
__global__ __launch_bounds__(256) void aggregate_kernel(
    const float* __restrict__ H, const float* __restrict__ As,
    const float* __restrict__ Ad, const int* __restrict__ off,
    const int* __restrict__ ebuf, const int* __restrict__ tgt,
    const int* __restrict__ ego, const float* __restrict__ gbias,
    float* __restrict__ feat) {
  const int g = blockIdx.x;
  __shared__ float s_as[NODES_PER_GRAPH][HEADS];
  __shared__ float s_ad[NODES_PER_GRAPH][HEADS];
  __shared__ float s_m[2][HEADS];
  __shared__ float s_den[2][HEADS];
  const int tid = threadIdx.x;
  if (tid < 128) {
    int i = tid >> 2, hd = tid & 3;
    s_as[i][hd] = As[(g * NODES_PER_GRAPH + i) * HEADS + hd];
  } else {
    int u = tid - 128;
    int i = u >> 2, hd = u & 3;
    s_ad[i][hd] = Ad[(g * NODES_PER_GRAPH + i) * HEADS + hd];
  }
  __syncthreads();
  const int e0 = off[g], e1 = off[g + 1];
  const int tgtL = tgt[g], egoL = ego[g];

  // deterministic serial max + denom per (slot, head): tiny edge counts
  if (tid < 8) {
    int s = tid >> 2, hd = tid & 3;
    int node = s ? egoL : tgtL;
    float m = -__builtin_inff();
    for (int e = e0; e < e1; ++e) {
      int p = ebuf[e];
      int dl = (p >> 8) & 31;
      if (dl != node) continue;
      int sl = p & 31;
      float ev = s_as[sl][hd] + s_ad[dl][hd];
      ev = ev > 0.f ? ev : NEG_SLOPE * ev;
      m = ev > m ? ev : m;
    }
    if (m == -__builtin_inff()) m = 0.f;   // reference: non-finite max -> 0
    float den = 0.f;
    for (int e = e0; e < e1; ++e) {
      int p = ebuf[e];
      int dl = (p >> 8) & 31;
      if (dl != node) continue;
      int sl = p & 31;
      float ev = s_as[sl][hd] + s_ad[dl][hd];
      ev = ev > 0.f ? ev : NEG_SLOPE * ev;
      den += __expf(ev - m);
    }
    s_m[s][hd] = m;
    s_den[s][hd] = den;
  }
  __syncthreads();

  // one thread per channel, loop over edges; H rows hit L2
  const int c = tid;
  const int hd = c >> 6;
  for (int s = 0; s < 2; ++s) {
    int node = s ? egoL : tgtL;
    float m = s_m[s][hd];
    float inv_den_add = s_den[s][hd] + 1e-16f;
    float acc = 0.f;
    for (int e = e0; e < e1; ++e) {
      int p = ebuf[e];
      int dl = (p >> 8) & 31;
      if (dl != node) continue;
      int sl = p & 31;
      float ev = s_as[sl][hd] + s_ad[dl][hd];
      ev = ev > 0.f ? ev : NEG_SLOPE * ev;
      float coef = __expf(ev - m) / inv_den_add;
      acc += H[(g * NODES_PER_GRAPH + sl) * EMB + c] * coef;
    }
    feat[g * (2 * EMB) + s * EMB + c] = acc + gbias[c];
  }
}

// =====================================================================
// K5: output = feat[4096,512] @ fc_W[512,256] + fc_b  (fp32 WMMA)
// =====================================================================
__global__ __launch_bounds__(256) void gemm_out_kernel(
    const float* __restrict__ F, const float* __restrict__ W,
    const float* __restrict__ B, float* __restrict__ O) {
  const int tid  = threadIdx.x;
  const int wave = tid >> 5, lane = tid & 31;
  const int rowTile = blockIdx.x * 2 + (wave >> 2);
  const int colQuad = wave & 3;
  const int row0 = rowTile * 16;
  const int col0 = colQuad * 64;
  const int half = lane >> 4;
  const int l15  = lane & 15;
  const int arow = row0 + l15;
  const int K = 2 * EMB;  // 512

  v8f acc0 = {}, acc1 = {}, acc2 = {}, acc3 = {};
  for (int k0 = 0; k0 < K; k0 += 4) {
    const int ka = k0 + half * 2;
    v2f a;
    a.x = F[arow * K + ka];
    a.y = F[arow * K + ka + 1];
    const float* Wk = W + ka * HIDDEN;
    v2f b0, b1, b2, b3;
    int c;
    c = col0 +  0 + l15; b0.x = Wk[c]; b0.y = Wk[HIDDEN + c];
    c = col0 + 16 + l15; b1.x = Wk[c]; b1.y = Wk[HIDDEN + c];
    c = col0 + 32 + l15; b2.x = Wk[c]; b2.y = Wk[HIDDEN + c];
    c = col0 + 48 + l15; b3.x = Wk[c]; b3.y = Wk[HIDDEN + c];
    acc0 = __builtin_amdgcn_wmma_f32_16x16x4_f32(false, a, false, b0, (short)0, acc0, false, false);
    acc1 = __builtin_amdgcn_wmma_f32_16x16x4_f32(false, a, false, b1, (short)0, acc1, false, false);
    acc2 = __builtin_amdgcn_wmma_f32_16x16x4_f32(false, a, false, b2, (short)0, acc2, false, false);
    acc3 = __builtin_amdgcn_wmma_f32_16x16x4_f32(false, a, false, b3, (short)0, acc3, false, false);
  }
  const int srow0 = row0 + half * 8;
#pragma unroll
  for (int r = 0; r < 8; ++r) {
    float* out = O + (srow0 + r) * HIDDEN + l15;
    out[col0 +  0] = acc0[r] + B[col0 +  0 + l15];
    out[col0 + 16] = acc1[r] + B[col0 + 16 + l15];
    out[col0 + 32] = acc2[r] + B[col0 + 32 + l15];
    out[col0 + 48] = acc3[r] + B[col0 + 48 + l15];
  }
}

// =====================================================================
extern "C" void kernel_launch(void* const* d_in, const int* in_sizes, int n_in,
                              void* d_out, int out_size, void* d_ws, size_t ws_size,
                              hipStream_t stream) {
  // inputs (setup_inputs order)
  const float* x       = (const float*)d_in[0];
  const int*   ei      = (const int*)d_in[1];   // [2, E] (src row, dst row)
  /* ptr d_in[2] unused: base = g*32 */
  const int*   tgt     = (const int*)d_in[3];
  const int*   ego     = (const int*)d_in[4];
  const float* W       = (const float*)d_in[5];
  const float* a_src   = (const float*)d_in[6];
  const float* a_dst   = (const float*)d_in[7];
  const float* gbias   = (const float*)d_in[8];
  const float* fc_W    = (const float*)d_in[9];
  const float* fc_b    = (const float*)d_in[10];
  float* out = (float*)d_out;

  // workspace partition (256B aligned)
  char* ws = (char*)d_ws;
  size_t o = 0;
  auto take = [&](size_t bytes) { char* p = ws + o; o = (o + bytes + 255) & ~(size_t)255; return p; };
  float* H      = (float*)take((size_t)NNODES * EMB * sizeof(float));      // 128 MB
  float* As     = (float*)take((size_t)NNODES * HEADS * sizeof(float));    // 2 MB
  float* Ad     = (float*)take((size_t)NNODES * HEADS * sizeof(float));    // 2 MB
  int*   cnt    = (int*)take((size_t)N_GRAPHS * sizeof(int));
  int*   off    = (int*)take((size_t)(N_GRAPHS + 1) * sizeof(int));
  int*   cursor = (int*)take((size_t)N_GRAPHS * sizeof(int));
  int*   ebuf   = (int*)take((size_t)NEDGES * sizeof(int));                // 4 MB worst case
  float* feat   = (float*)take((size_t)N_GRAPHS * 2 * EMB * sizeof(float)); // 8 MB

  hipMemsetAsync(cnt, 0, (size_t)N_GRAPHS * sizeof(int), stream);

  // K1: h = x @ W   (32 rows x 256 cols per block)
  gemm_xw_kernel<<<NNODES / 32, 256, 0, stream>>>(x, W, H);
  // K2: attention logits per node/head
  alphas_kernel<<<(NNODES * HEADS + 255) / 256, 256, 0, stream>>>(H, a_src, a_dst, As, Ad);
  // K3: bin edges whose dst is the target/ego node of its graph
  edge_hist_kernel<<<NEDGES / 256, 256, 0, stream>>>(ei, tgt, ego, cnt);
  scan_kernel<<<1, 256, 0, stream>>>(cnt, off, cursor);
  edge_scatter_kernel<<<NEDGES / 256, 256, 0, stream>>>(ei, tgt, ego, cursor, ebuf);
  // K4: per-graph softmax + aggregation -> feat
  aggregate_kernel<<<N_GRAPHS, 256, 0, stream>>>(H, As, Ad, off, ebuf, tgt, ego, gbias, feat);
  // K5: output = feat @ fc_W + fc_b
  gemm_out_kernel<<<N_GRAPHS / 32, 256, 0, stream>>>(feat, fc_W, fc_b, out);
}